// RGNN_42795054137552
// MI455X (gfx1250) — compile-verified
//
#include <hip/hip_runtime.h>

typedef float v2f __attribute__((ext_vector_type(2)));
typedef float v8f __attribute__((ext_vector_type(8)));

constexpr int N_ = 20000, E_ = 320000, G_ = 64;
constexpr int NH = 128, M_ = 4;
constexpr float EPSB = 1e-5f;

// ---------------- row moment stats: S1[16] = sum rows, S2[256] = sum a a^T ----
__global__ __launch_bounds__(256) void k_rowstats(const float* __restrict__ A, int R,
                                                  float* __restrict__ S1, float* __restrict__ S2) {
  __shared__ float buf[64][16];
  int t = threadIdx.x;
  int i = t >> 4, j = t & 15;
  float s2 = 0.f, s1 = 0.f;
  for (int base = blockIdx.x * 64; base < R; base += gridDim.x * 64) {
    int r = base + (t >> 2);
    int c = (t & 3) * 4;
    float4 v = make_float4(0.f, 0.f, 0.f, 0.f);
    if (r < R) v = *(const float4*)(A + (size_t)r * 16 + c);
    buf[t >> 2][c + 0] = v.x; buf[t >> 2][c + 1] = v.y;
    buf[t >> 2][c + 2] = v.z; buf[t >> 2][c + 3] = v.w;
    __syncthreads();
#pragma unroll 8
    for (int rr = 0; rr < 64; ++rr) {
      s2 += buf[rr][i] * buf[rr][j];
      if (t < 16) s1 += buf[rr][t];
    }
    __syncthreads();
  }
  atomicAdd(&S2[t], s2);
  if (t < 16) atomicAdd(&S1[t], s1);
}

// ---------------- fold BN (analytic batch stats) into a 16->128 linear --------
__global__ void k_fold(const float* __restrict__ W, const float* __restrict__ b,
                       const float* __restrict__ g, const float* __restrict__ bt,
                       const float* __restrict__ S1, const float* __restrict__ S2,
                       float invR, float* __restrict__ Wf, float* __restrict__ tf) {
  int c = threadIdx.x;
  if (c >= NH) return;
  float w[16];
#pragma unroll
  for (int k = 0; k < 16; ++k) w[k] = W[c * 16 + k];
  float wS1 = 0.f;
#pragma unroll
  for (int k = 0; k < 16; ++k) wS1 += w[k] * S1[k];
  float q = 0.f;
  for (int a = 0; a < 16; ++a) {
    float wa = w[a];
#pragma unroll
    for (int d = 0; d < 16; ++d) q += wa * S2[a * 16 + d] * w[d];
  }
  float bc = b[c];
  float mean = wS1 * invR + bc;
  float msq = q * invR + 2.f * bc * wS1 * invR + bc * bc;
  float var = msq - mean * mean;
  float s = g[c] * rsqrtf(var + EPSB);
#pragma unroll
  for (int k = 0; k < 16; ++k) Wf[c * 16 + k] = w[k] * s;
  tf[c] = (bc - mean) * s + bt[c];
}

// ---------------- input encoder: h = relu(folded linear) ---------------------
__global__ __launch_bounds__(256) void k_h(const float* __restrict__ xn, const float* __restrict__ Wf,
                                           const float* __restrict__ tf, float* __restrict__ h) {
  int idx = blockIdx.x * 256 + threadIdx.x;
  if (idx >= N_ * NH) return;
  int n = idx >> 7, c = idx & 127;
  const float4* a = (const float4*)(xn + (size_t)n * 16);
  const float4* w = (const float4*)(Wf + (size_t)c * 16);
  float acc = tf[c];
#pragma unroll
  for (int k = 0; k < 4; ++k) {
    float4 av = a[k], wv = w[k];
    acc += av.x * wv.x + av.y * wv.y + av.z * wv.z + av.w * wv.w;
  }
  h[idx] = fmaxf(acc, 0.f);
}

// ---------------- WMMA GEMM: D(R,128) = A(R,128) @ W(128,128)^T --------------
// MODE 0: A rows contiguous (stride 128), D rows contiguous (stride 128)   [Linear2 main]
// MODE 1: A & D in (n, c, m) layout, logical row r = 4n + m                [conv einsum]
template <int MODE>
__global__ __launch_bounds__(256) void k_gemm(const float* __restrict__ Ag, const float* __restrict__ Wg,
                                              int ldb, float* __restrict__ Dg) {
  __shared__ float Alds[16 * 132];   // 16 rows x 128 (pad 132) : 8.25 KB
  __shared__ float Wl[128 * 68];     // 128 rows x 64-k half (pad 68) : 34 KB
  int t = threadIdx.x;

  // stage A tile (2048 floats, fully coalesced float4 reads)
  const float4* src = (const float4*)(Ag + (size_t)blockIdx.x * 2048);
#pragma unroll
  for (int i = 0; i < 2; ++i) {
    int fi = t + i * 256;           // 0..511 float4
    float4 v = src[fi];
    if (MODE == 0) {
      int r = fi >> 5, c4 = fi & 31;                 // 32 float4 per row
      *(float4*)(&Alds[r * 132 + c4 * 4]) = v;
    } else {
      int dn = fi >> 7, c = fi & 127;                // node-local, channel; comps = m
      int rb = dn * 4;
      Alds[(rb + 0) * 132 + c] = v.x;
      Alds[(rb + 1) * 132 + c] = v.y;
      Alds[(rb + 2) * 132 + c] = v.z;
      Alds[(rb + 3) * 132 + c] = v.w;
    }
  }

  int lane = t & 31, wv = t >> 5;
  int rr = lane & 15, hi = lane >> 4;
  int o0 = wv * 16;
  v8f acc = {0.f, 0.f, 0.f, 0.f, 0.f, 0.f, 0.f, 0.f};

  for (int ph = 0; ph < 2; ++ph) {
    // stage 64-deep half of W (B^T), row-major with pad 68
    for (int idx = t; idx < 128 * 64; idx += 256) {
      int o = idx >> 6, k = idx & 63;
      Wl[o * 68 + k] = Wg[(size_t)o * ldb + ph * 64 + k];
    }
    __syncthreads();
#pragma unroll
    for (int k0 = 0; k0 < 64; k0 += 4) {
      int kA = k0 + 2 * hi;
      v2f a = *(const v2f*)(&Alds[rr * 132 + ph * 64 + kA]);
      v2f b = *(const v2f*)(&Wl[(o0 + rr) * 68 + kA]);
      acc = __builtin_amdgcn_wmma_f32_16x16x4_f32(false, a, false, b, (short)0, acc, false, false);
    }
    __syncthreads();
  }

  int row0 = blockIdx.x * 16;
#pragma unroll
  for (int v = 0; v < 8; ++v) {
    int r = row0 + v + hi * 8;      // D: VGPR v -> M = v (+8 for upper lanes), N = lane%16
    int col = o0 + rr;
    if (MODE == 0) {
      Dg[(size_t)r * 128 + col] = acc[v];
    } else {
      int n = r >> 2, m = r & 3;
      Dg[(size_t)n * 512 + col * 4 + m] = acc[v];
    }
  }
}

// ---------------- Linear2 rand-term scramble + broadcast ---------------------
__global__ __launch_bounds__(256) void k_x_init(const float* __restrict__ mainb, const float* __restrict__ ax,
                                                const float* __restrict__ W_l2, const float* __restrict__ b_l2,
                                                float* __restrict__ x) {
  int idx = blockIdx.x * 256 + threadIdx.x;
  if (idx >= N_ * 512) return;
  int n = idx >> 9, rem = idx & 511;
  int c = rem >> 2;
  int c2 = rem & 127, j = rem >> 7;        // rem = 4c + m
  int i = 4 * n + j;
  int p = i % N_, q = i / N_;
  float xr = ax[(size_t)p * 4 + q];
  x[idx] = mainb[(size_t)n * 128 + c] + b_l2[c2] + xr * W_l2[(size_t)c2 * 129 + 128];
}

// ---------------- message + aggregate (edge encoder fused, folded BN) --------
__global__ __launch_bounds__(256) void k_msg(const float* __restrict__ ea, const int* __restrict__ ei,
                                             const float* __restrict__ Wf_g, const float* __restrict__ tf,
                                             const float* __restrict__ x, float* __restrict__ agg) {
  __shared__ float WfT[16 * 128];   // transposed [k][c] for conflict-free reads
  __shared__ float tfl[128];
  int t = threadIdx.x;
  for (int idx = t; idx < 2048; idx += 256) {
    int c = idx >> 4, k = idx & 15;
    WfT[k * 128 + c] = Wf_g[idx];
  }
  if (t < 128) tfl[t] = tf[t];
  __syncthreads();

  int wv = t >> 5, lane = t & 31;
  int e = blockIdx.x * 8 + wv;
  if (e >= E_) return;
  int src = ei[e], dst = ei[E_ + e];

  float a0[16];
  const float4* ap = (const float4*)(ea + (size_t)e * 16);
#pragma unroll
  for (int k4 = 0; k4 < 4; ++k4) {
    float4 v = ap[k4];
    a0[k4 * 4 + 0] = v.x; a0[k4 * 4 + 1] = v.y; a0[k4 * 4 + 2] = v.z; a0[k4 * 4 + 3] = v.w;
  }
  size_t bs = (size_t)src * 512, bd = (size_t)dst * 512;
#pragma unroll
  for (int j = 0; j < 4; ++j) {
    int c = lane + 32 * j;                         // coalesced float4 across wave
    float ec = tfl[c];
#pragma unroll
    for (int k = 0; k < 16; ++k) ec += a0[k] * WfT[k * 128 + c];
    ec = fmaxf(ec, 0.f);
    float4 xv = *(const float4*)(x + bs + c * 4);
    atomicAdd(&agg[bd + c * 4 + 0], fmaxf(xv.x + ec, 0.f));
    atomicAdd(&agg[bd + c * 4 + 1], fmaxf(xv.y + ec, 0.f));
    atomicAdd(&agg[bd + c * 4 + 2], fmaxf(xv.z + ec, 0.f));
    atomicAdd(&agg[bd + c * 4 + 3], fmaxf(xv.w + ec, 0.f));
  }
}

// ---------------- per-channel stats of h3 over (n, m) ------------------------
__global__ void k_ch_stats(const float* __restrict__ H3, float* __restrict__ st) {
  int o = threadIdx.x;   // 128
  float s = 0.f, sq = 0.f;
  for (int n = blockIdx.x; n < N_; n += gridDim.x) {
    float4 v = *(const float4*)(H3 + (size_t)n * 512 + o * 4);
    s += v.x + v.y + v.z + v.w;
    sq += v.x * v.x + v.y * v.y + v.z * v.z + v.w * v.w;
  }
  atomicAdd(&st[o], s);
  atomicAdd(&st[128 + o], sq);
}

// ---------------- BN3 + relu + residual (in place on x) ----------------------
__global__ __launch_bounds__(256) void k_bnrelu_res(const float* __restrict__ H3, const float* __restrict__ st,
                                                    const float* __restrict__ g, const float* __restrict__ bt,
                                                    float* __restrict__ x) {
  int idx = blockIdx.x * 256 + threadIdx.x;
  if (idx >= N_ * 512) return;
  int o = (idx >> 2) & 127;
  const float inv = 1.f / (float)(N_ * M_);
  float mu = st[o] * inv;
  float var = st[128 + o] * inv - mu * mu;
  float v = (H3[idx] - mu) * rsqrtf(var + EPSB) * g[o] + bt[o];
  x[idx] += fmaxf(v, 0.f);
}

// ---------------- graph add-pool ---------------------------------------------
__global__ __launch_bounds__(256) void k_pool(const float* __restrict__ x, const int* __restrict__ batch,
                                              float* __restrict__ xg) {
  int idx = blockIdx.x * 256 + threadIdx.x;
  if (idx >= N_ * 512) return;
  int n = idx >> 9;
  int gph = batch[n];
  atomicAdd(&xg[(size_t)gph * 512 + (idx & 511)], x[idx]);
}

// ---------------- output head (256 rows, one block) --------------------------
__global__ __launch_bounds__(256) void k_out(const float* __restrict__ xg,
                                             const float* __restrict__ W1, const float* __restrict__ b1,
                                             const float* __restrict__ gg, const float* __restrict__ btg,
                                             const float* __restrict__ W2, const float* __restrict__ b2,
                                             float* __restrict__ out) {
  __shared__ float ssum[64], ssq[64], sc[64], sh[64];
  int r = threadIdx.x;   // 256 rows of flat = xg viewed (256,128)
  float acc[64];
#pragma unroll
  for (int jj = 0; jj < 64; ++jj) acc[jj] = b1[jj];
  for (int c = 0; c < 128; ++c) {
    float fr = xg[(size_t)r * 128 + c];
#pragma unroll
    for (int jj = 0; jj < 64; ++jj) acc[jj] += fr * W1[jj * 128 + c];
  }
  if (r < 64) { ssum[r] = 0.f; ssq[r] = 0.f; }
  __syncthreads();
#pragma unroll
  for (int jj = 0; jj < 64; ++jj) {
    atomicAdd(&ssum[jj], acc[jj]);
    atomicAdd(&ssq[jj], acc[jj] * acc[jj]);
  }
  __syncthreads();
  if (r < 64) {
    float mu = ssum[r] * (1.f / 256.f);
    float var = ssq[r] * (1.f / 256.f) - mu * mu;
    float s = gg[r] * rsqrtf(var + EPSB);
    sc[r] = s;
    sh[r] = btg[r] - mu * s;
  }
  __syncthreads();
  float o1[64];
#pragma unroll
  for (int jj = 0; jj < 64; ++jj) o1[jj] = fmaxf(acc[jj] * sc[jj] + sh[jj], 0.f);

  int q = r >> 2, s4 = r & 3;
  float res[16];
#pragma unroll
  for (int t2 = 0; t2 < 16; ++t2) res[t2] = 0.f;
  for (int tt = 0; tt < 64; ++tt) {
    float v = b2[tt];
#pragma unroll
    for (int c = 0; c < 64; ++c) v += o1[c] * W2[tt * 64 + c];
    res[tt >> 2] += v;
  }
#pragma unroll
  for (int t2 = 0; t2 < 16; ++t2) out[(size_t)q * 64 + 16 * s4 + t2] = 0.25f * res[t2];
}

// ============================================================================
extern "C" void kernel_launch(void* const* d_in, const int* in_sizes, int n_in,
                              void* d_out, int out_size, void* d_ws, size_t ws_size,
                              hipStream_t stream) {
  (void)in_sizes; (void)n_in; (void)out_size; (void)ws_size;
  const float* x_node   = (const float*)d_in[0];
  const float* edge_attr= (const float*)d_in[1];
  const int*   edge_idx = (const int*)d_in[2];
  const int*   batch    = (const int*)d_in[3];
  const float* ax       = (const float*)d_in[4];
  const float* W_in     = (const float*)d_in[5];
  const float* b_in     = (const float*)d_in[6];
  const float* g_in     = (const float*)d_in[7];
  const float* bt_in    = (const float*)d_in[8];
  const float* W_l2     = (const float*)d_in[9];
  const float* b_l2     = (const float*)d_in[10];
  const float* W_edge   = (const float*)d_in[11];
  const float* b_edge   = (const float*)d_in[12];
  const float* g_edge   = (const float*)d_in[13];
  const float* bt_edge  = (const float*)d_in[14];
  const float* W_conv   = (const float*)d_in[15];
  const float* g_norm   = (const float*)d_in[16];
  const float* bt_norm  = (const float*)d_in[17];
  const float* W_out1   = (const float*)d_in[18];
  const float* b_out1   = (const float*)d_in[19];
  const float* g_out    = (const float*)d_in[20];
  const float* bt_out   = (const float*)d_in[21];
  const float* W_out2   = (const float*)d_in[22];
  const float* b_out2   = (const float*)d_in[23];
  float* out = (float*)d_out;

  float* ws = (float*)d_ws;
  size_t off = 0;
  auto alloc = [&](size_t n) { float* p = ws + off; off += (n + 63) & ~size_t(63); return p; };
  float* h     = alloc((size_t)N_ * NH);
  float* mainb = alloc((size_t)N_ * NH);
  float* x     = alloc((size_t)N_ * NH * M_);
  float* agg   = alloc((size_t)N_ * NH * M_);
  float* h3    = alloc((size_t)N_ * NH * M_);
  float* S1n = alloc(16);  float* S2n = alloc(256);
  float* S1e = alloc(16);  float* S2e = alloc(256);
  float* Wf_in = alloc(NH * 16);     float* tf_in = alloc(NH);
  float* Wf_e  = alloc(3 * NH * 16); float* tf_e  = alloc(3 * NH);
  float* st = alloc(256);
  float* xg = alloc((size_t)G_ * NH * M_);

  hipMemsetAsync(S1n, 0, 16 * sizeof(float), stream);
  hipMemsetAsync(S2n, 0, 256 * sizeof(float), stream);
  hipMemsetAsync(S1e, 0, 16 * sizeof(float), stream);
  hipMemsetAsync(S2e, 0, 256 * sizeof(float), stream);

  k_rowstats<<<320, 256, 0, stream>>>(x_node, N_, S1n, S2n);
  k_rowstats<<<512, 256, 0, stream>>>(edge_attr, E_, S1e, S2e);
  k_fold<<<1, 128, 0, stream>>>(W_in, b_in, g_in, bt_in, S1n, S2n, 1.f / (float)N_, Wf_in, tf_in);
  for (int l = 0; l < 3; ++l)
    k_fold<<<1, 128, 0, stream>>>(W_edge + l * 2048, b_edge + l * 128, g_edge + l * 128,
                                  bt_edge + l * 128, S1e, S2e, 1.f / (float)E_,
                                  Wf_e + l * 2048, tf_e + l * 128);

  k_h<<<(N_ * NH) / 256, 256, 0, stream>>>(x_node, Wf_in, tf_in, h);
  k_gemm<0><<<N_ / 16, 256, 0, stream>>>(h, W_l2, 129, mainb);
  k_x_init<<<(N_ * 512) / 256, 256, 0, stream>>>(mainb, ax, W_l2, b_l2, x);

  for (int l = 0; l < 3; ++l) {
    hipMemcpyAsync(agg, x, sizeof(float) * (size_t)N_ * 512, hipMemcpyDeviceToDevice, stream);
    k_msg<<<E_ / 8, 256, 0, stream>>>(edge_attr, edge_idx, Wf_e + l * 2048, tf_e + l * 128, x, agg);
    k_gemm<1><<<(N_ * M_) / 16, 256, 0, stream>>>(agg, W_conv + (size_t)l * NH * NH, 128, h3);
    hipMemsetAsync(st, 0, 256 * sizeof(float), stream);
    k_ch_stats<<<256, 128, 0, stream>>>(h3, st);
    k_bnrelu_res<<<(N_ * 512) / 256, 256, 0, stream>>>(h3, st, g_norm + l * 128, bt_norm + l * 128, x);
  }

  hipMemsetAsync(xg, 0, sizeof(float) * (size_t)G_ * 512, stream);
  k_pool<<<(N_ * 512) / 256, 256, 0, stream>>>(x, batch, xg);
  k_out<<<1, 256, 0, stream>>>(xg, W_out1, b_out1, g_out, bt_out, W_out2, b_out2, out);
}